// BiLSTM_CRF_87376814670214
// MI455X (gfx1250) — compile-verified
//
#include <hip/hip_runtime.h>
#include <hip/hip_bf16.h>

// ---------------- problem constants ----------------
#define VOCAB 100000
#define E     512
#define HID   1024
#define H     512          // HID/2
#define G4    2048         // 4*H gate rows
#define Lseq  4096
#define T     7
#define START 4
#define STOP  5
#define NEGV  (-10000.0f)

// recurrence partitioning
#define NWG_DIR 128        // workgroups per direction
#define HPW     4          // h elements owned per workgroup (H / NWG_DIR)

// GEMM tiling
#define BM   64
#define BN   128
#define BK   64
#define LPAD 66            // padded LDS row stride (floats): banks spread, 8B aligned

typedef __attribute__((ext_vector_type(2))) float v2f;
typedef __attribute__((ext_vector_type(8))) float v8f;
typedef __attribute__((ext_vector_type(4))) int   v4i;

// pointer-to-AS1(v4i) and pointer-to-AS3(v4i), matching the async builtin's params
typedef __attribute__((address_space(1))) v4i* as1_v4i;
typedef __attribute__((address_space(3))) v4i* as3_v4i;

// CDNA5 async global->LDS staging (ASYNCcnt-tracked), with safe fallback.
#if defined(__has_builtin)
#if __has_builtin(__builtin_amdgcn_global_load_async_to_lds_b128)
#define HAVE_ASYNC_LDS 1
#endif
#endif

#ifdef HAVE_ASYNC_LDS
#define ASYNC_COPY16(ldsp, gp)                                                  \
    __builtin_amdgcn_global_load_async_to_lds_b128(                             \
        (as1_v4i)(unsigned long long)(gp),                                      \
        (as3_v4i)(unsigned)(unsigned long long)(ldsp), 0, 0)
#if __has_builtin(__builtin_amdgcn_s_wait_asynccnt)
#define ASYNC_WAIT() __builtin_amdgcn_s_wait_asynccnt(0)
#else
#define ASYNC_WAIT() asm volatile("s_wait_asynccnt 0" ::: "memory")
#endif
#else
#define ASYNC_COPY16(ldsp, gp) (*(float4*)(ldsp) = *(const float4*)(gp))
#define ASYNC_WAIT()
#endif

// ---------------- workspace layout (floats unless noted) ----------------
#define X_OFF    ((size_t)0)
#define XGF_OFF  (X_OFF   + (size_t)Lseq * E)
#define XGB_OFF  (XGF_OFF + (size_t)Lseq * G4)
#define HF_OFF   (XGB_OFF + (size_t)Lseq * G4)
#define HB_OFF   (HF_OFF  + (size_t)Lseq * H)
#define FEAT_OFF (HB_OFF  + (size_t)Lseq * H)
#define BPTR_OFF (FEAT_OFF+ (size_t)Lseq * T)
#define BAR_OFF  (BPTR_OFF+ (size_t)Lseq * T)

// ---------------- kernel 0: zero the barrier counters ----------------
__global__ void zero_bar_kernel(int* __restrict__ bar) {
    int i = blockIdx.x * blockDim.x + threadIdx.x;
    if (i < 2 * Lseq) bar[i] = 0;
}

// ---------------- kernel 1: embedding gather ----------------
__global__ __launch_bounds__(256) void gather_kernel(const int* __restrict__ sent,
                                                     const float* __restrict__ emb,
                                                     float* __restrict__ x) {
    int i   = blockIdx.x * blockDim.x + threadIdx.x;   // float4 index
    int row = i >> 7;                                  // / (E/4)
    int col = (i & 127) << 2;
    int tok = sent[row];
    *(float4*)(x + (size_t)row * E + col) =
        *(const float4*)(emb + (size_t)tok * E + col);
}

// ---------------- kernel 2: input projection GEMM via f32 WMMA ----------------
// xg[t][n] = sum_e x[t][e] * w_ih[n][e] + b_ih[n] + b_hh[n]
// Block: 64(M) x 128(N) tile; 8 waves, each a 32x32 tile = 2x2 WMMA sub-tiles.
// A/B tiles staged to LDS via async global->LDS b128 copies, rows padded to
// LPAD floats for bank spread. Per K-step of 4: 4 ds_load_b64 -> 4 WMMAs.
__global__ __launch_bounds__(256) void gemm_xg_kernel(const float* __restrict__ x,
                                                      const float* __restrict__ w_f,
                                                      const float* __restrict__ w_b,
                                                      const float* __restrict__ bih_f,
                                                      const float* __restrict__ bhh_f,
                                                      const float* __restrict__ bih_b,
                                                      const float* __restrict__ bhh_b,
                                                      float* __restrict__ xg_f,
                                                      float* __restrict__ xg_b) {
    const float* w   = blockIdx.z ? w_b   : w_f;
    const float* bih = blockIdx.z ? bih_b : bih_f;
    const float* bhh = blockIdx.z ? bhh_b : bhh_f;
    float*       xg  = blockIdx.z ? xg_b  : xg_f;

    __shared__ float Atile[BM * LPAD];   // ~16.9 KB
    __shared__ float Btile[BN * LPAD];   // ~33.8 KB

    int m0g = blockIdx.x * BM;
    int n0g = blockIdx.y * BN;

    int tid     = threadIdx.x;
    int lane    = tid & 31;
    int wv      = tid >> 5;              // 0..7
    int half    = lane >> 4;             // 0: K pair {0,1}, 1: K pair {2,3}
    int lanemod = lane & 15;
    int ml      = (wv >> 2) << 5;        // wave M offset: 0 or 32
    int nl      = (wv & 3) << 5;         // wave N offset: 0,32,64,96

    // loader assignments (16B chunks)
    int arow   = tid >> 2;               // 0..63
    int achunk = (tid & 3) << 2;         // 0,4,8,12 floats

    v8f acc00 = {}, acc01 = {}, acc10 = {}, acc11 = {};

    for (int kb = 0; kb < E; kb += BK) {
        // A tile: 64 rows x 16 floats (cols 0..15) = 256 x 16B chunks, 1 per thread
        ASYNC_COPY16(&Atile[arow * LPAD + achunk],
                     x + (size_t)(m0g + arow) * E + kb + achunk);
        // B tile: 128 rows x 16 floats (cols 0..15) = 512 x 16B chunks, 2 per thread
        {
            int c0 = tid, c1 = tid + 256;
            int r0 = c0 >> 2, ch0 = (c0 & 3) << 2;
            int r1 = c1 >> 2, ch1 = (c1 & 3) << 2;
            ASYNC_COPY16(&Btile[r0 * LPAD + ch0],
                         w + (size_t)(n0g + r0) * E + kb + ch0);
            ASYNC_COPY16(&Btile[r1 * LPAD + ch1],
                         w + (size_t)(n0g + r1) * E + kb + ch1);
        }
        ASYNC_WAIT();
        __syncthreads();

        const float* ap0 = &Atile[(ml + lanemod) * LPAD + 2 * half];
        const float* ap1 = ap0 + 16 * LPAD;
        const float* bp0 = &Btile[(nl + lanemod) * LPAD + 2 * half];
        const float* bp1 = bp0 + 16 * LPAD;

#pragma unroll 4
        for (int kk = 0; kk < BK; kk += 4) {
            v2f a0 = *(const v2f*)(ap0 + kk);
            v2f a1 = *(const v2f*)(ap1 + kk);
            v2f b0 = *(const v2f*)(bp0 + kk);
            v2f b1 = *(const v2f*)(bp1 + kk);
            acc00 = __builtin_amdgcn_wmma_f32_16x16x4_f32(false, a0, false, b0,
                                                          (short)0, acc00, false, false);
            acc01 = __builtin_amdgcn_wmma_f32_16x16x4_f32(false, a0, false, b1,
                                                          (short)0, acc01, false, false);
            acc10 = __builtin_amdgcn_wmma_f32_16x16x4_f32(false, a1, false, b0,
                                                          (short)0, acc10, false, false);
            acc11 = __builtin_amdgcn_wmma_f32_16x16x4_f32(false, a1, false, b1,
                                                          (short)0, acc11, false, false);
        }
        __syncthreads();
    }

    // epilogue: D row m = base + r + 8*half, col n = base + lanemod
    int nn0 = n0g + nl + lanemod;
    int nn1 = nn0 + 16;
    float bias0 = bih[nn0] + bhh[nn0];
    float bias1 = bih[nn1] + bhh[nn1];
#pragma unroll
    for (int r = 0; r < 8; ++r) {
        int mm0 = m0g + ml + r + 8 * half;
        int mm1 = mm0 + 16;
        xg[(size_t)mm0 * G4 + nn0] = acc00[r] + bias0;
        xg[(size_t)mm0 * G4 + nn1] = acc01[r] + bias1;
        xg[(size_t)mm1 * G4 + nn0] = acc10[r] + bias0;
        xg[(size_t)mm1 * G4 + nn1] = acc11[r] + bias1;
    }
}

// ---------------- kernel 3: persistent bidirectional LSTM recurrence ----------------
__global__ __launch_bounds__(256) void lstm_rec_kernel(const float* __restrict__ xg_f,
                                                       const float* __restrict__ xg_b,
                                                       const float* __restrict__ whh_f,
                                                       const float* __restrict__ whh_b,
                                                       float* __restrict__ h_f,
                                                       float* __restrict__ h_b,
                                                       int* __restrict__ bar) {
    int dir = blockIdx.x / NWG_DIR;
    int wg  = blockIdx.x % NWG_DIR;
    const float* xg  = dir ? xg_b  : xg_f;
    const float* whh = dir ? whh_b : whh_f;
    float*       hout= dir ? h_b   : h_f;
    int*         cnt = bar + dir * Lseq;

    __shared__ float wlds[16 * H];     // 32 KB: 16 gate rows x 512
    __shared__ float gred[16];
    __shared__ float c_state[HPW];

    int tid = threadIdx.x;
    int j0  = wg * HPW;                // first owned h index
    for (int i = tid; i < 16 * H; i += 256) {
        int r = i >> 9;
        int k = i & (H - 1);
        int q = r >> 2, e = r & 3;
        wlds[i] = whh[(size_t)(q * H + j0 + e) * H + k];
    }
    if (tid < HPW) c_state[tid] = 0.0f;
    __syncthreads();

    int row = tid >> 4;                // 0..15 local gate row
    int sub = tid & 15;                // 16 lanes per row

    for (int s = 0; s < Lseq; ++s) {
        int t = dir ? (Lseq - 1 - s) : s;
        float partial = 0.0f;
        if (s > 0) {
            int tp = dir ? (t + 1) : (t - 1);
            const float* hp = hout + (size_t)tp * H;
            const float* wr = wlds + row * H + sub * 32;
#pragma unroll 8
            for (int kk = 0; kk < 32; ++kk)
                partial = fmaf(wr[kk], hp[sub * 32 + kk], partial);
        }
#pragma unroll
        for (int off = 1; off < 16; off <<= 1)
            partial += __shfl_xor(partial, off, 16);
        if (sub == 0) gred[row] = partial;

        // prefetch next timestep's xg gate slice while we sync/spin
        if (tid == 32 && s + 1 < Lseq) {
            int tn = dir ? (t - 1) : (t + 1);
            __builtin_prefetch(xg + (size_t)tn * G4 + j0, 0, 1);
        }
        __syncthreads();

        if (tid < HPW) {
            int e = tid;
            int j = j0 + e;
            const float* xgt = xg + (size_t)t * G4;
            float gi = xgt[0 * H + j] + gred[0 + e];
            float gf = xgt[1 * H + j] + gred[4 + e];
            float gg = xgt[2 * H + j] + gred[8 + e];
            float go = xgt[3 * H + j] + gred[12 + e];
            float i_ = 1.0f / (1.0f + __expf(-gi));
            float f_ = 1.0f / (1.0f + __expf(-gf));
            float g_ = tanhf(gg);
            float o_ = 1.0f / (1.0f + __expf(-go));
            float c  = f_ * c_state[e] + i_ * g_;
            c_state[e] = c;
            hout[(size_t)t * H + j] = o_ * tanhf(c);
        }
        __syncthreads();

        if (tid == 0) {
            __threadfence();                               // publish h
            __hip_atomic_fetch_add(cnt + s, 1, __ATOMIC_RELEASE,
                                   __HIP_MEMORY_SCOPE_AGENT);
            while (__hip_atomic_load(cnt + s, __ATOMIC_ACQUIRE,
                                     __HIP_MEMORY_SCOPE_AGENT) < NWG_DIR) {
                __builtin_amdgcn_s_sleep(1);
            }
        }
        __syncthreads();
    }
}

// ---------------- kernel 4: output projection (feats) ----------------
__global__ __launch_bounds__(224) void feats_kernel(const float* __restrict__ hf,
                                                    const float* __restrict__ hb,
                                                    const float* __restrict__ w_out,
                                                    const float* __restrict__ b_out,
                                                    float* __restrict__ feats) {
    int t    = blockIdx.x;
    int tag  = threadIdx.x >> 5;
    int lane = threadIdx.x & 31;
    if (tag >= T) return;
    const float* hfr = hf + (size_t)t * H;
    const float* hbr = hb + (size_t)t * H;
    const float* wf  = w_out + (size_t)tag * HID;
    const float* wb  = wf + H;
    float s = 0.0f;
    for (int k = lane; k < H; k += 32) {
        s = fmaf(hfr[k], wf[k], s);
        s = fmaf(hbr[k], wb[k], s);
    }
#pragma unroll
    for (int off = 16; off; off >>= 1) s += __shfl_xor(s, off, 32);
    if (lane == 0) feats[(size_t)t * T + tag] = s + b_out[tag];
}

// ---------------- kernel 5: Viterbi forward + backtrace ----------------
__global__ __launch_bounds__(32) void viterbi_kernel(const float* __restrict__ feats,
                                                     const float* __restrict__ trans,
                                                     int* __restrict__ bptrs,
                                                     void* __restrict__ out) {
    __shared__ float fv[T], fv2[T];
    __shared__ float tr[T * T];
    int tid = threadIdx.x;
    if (tid < T * T) tr[tid] = trans[tid];
    if (tid < T) fv[tid] = (tid == START) ? 0.0f : NEGV;
    __syncthreads();

    for (int t = 0; t < Lseq; ++t) {
        if (tid < T) {
            float best = -3.4e38f; int bi = 0;
#pragma unroll
            for (int s = 0; s < T; ++s) {
                float v = fv[s] + tr[tid * T + s];
                if (v > best) { best = v; bi = s; }
            }
            bptrs[(size_t)t * T + tid] = bi;
            fv2[tid] = best + feats[(size_t)t * T + tid];
        }
        __syncthreads();
        if (tid < T) fv[tid] = fv2[tid];
        __syncthreads();
    }

    if (tid == 0) {
        float best = -3.4e38f; int btag = 0;
#pragma unroll
        for (int s = 0; s < T; ++s) {
            float v = fv[s] + tr[STOP * T + s];
            if (v > best) { best = v; btag = s; }
        }
        int*   path = (int*)out;
        float* fout = (float*)out;
        int tag = btag;
        for (int t = Lseq - 1; t >= 0; --t) {
            path[t] = tag;
            tag = bptrs[(size_t)t * T + tag];
        }
        fout[Lseq] = best;   // score
    }
}

// ---------------- launch ----------------
extern "C" void kernel_launch(void* const* d_in, const int* in_sizes, int n_in,
                              void* d_out, int out_size, void* d_ws, size_t ws_size,
                              hipStream_t stream) {
    const int*   sentence = (const int*)  d_in[0];
    const float* emb      = (const float*)d_in[1];
    const float* w_ih_f   = (const float*)d_in[2];
    const float* w_hh_f   = (const float*)d_in[3];
    const float* b_ih_f   = (const float*)d_in[4];
    const float* b_hh_f   = (const float*)d_in[5];
    const float* w_ih_b   = (const float*)d_in[6];
    const float* w_hh_b   = (const float*)d_in[7];
    const float* b_ih_b   = (const float*)d_in[8];
    const float* b_hh_b   = (const float*)d_in[9];
    const float* w_out    = (const float*)d_in[10];
    const float* b_out    = (const float*)d_in[11];
    const float* trans    = (const float*)d_in[12];

    float* ws    = (float*)d_ws;
    float* x     = ws + X_OFF;
    float* xg_f  = ws + XGF_OFF;
    float* xg_b  = ws + XGB_OFF;
    float* h_f   = ws + HF_OFF;
    float* h_b   = ws + HB_OFF;
    float* feats = ws + FEAT_OFF;
    int*   bptrs = (int*)(ws + BPTR_OFF);
    int*   bar   = (int*)(ws + BAR_OFF);

    // 0) barrier counters must be zero every call (deterministic replay)
    zero_bar_kernel<<<(2 * Lseq + 255) / 256, 256, 0, stream>>>(bar);

    // 1) gather embeddings
    gather_kernel<<<(Lseq * E / 4) / 256, 256, 0, stream>>>(sentence, emb, x);

    // 2) input projections, both directions, f32 WMMA with async LDS staging
    {
        dim3 grid(Lseq / BM, G4 / BN, 2);   // 64 x 16 x 2
        gemm_xg_kernel<<<grid, 256, 0, stream>>>(x, w_ih_f, w_ih_b,
                                                 b_ih_f, b_hh_f, b_ih_b, b_hh_b,
                                                 xg_f, xg_b);
    }

    // 3) persistent bidirectional recurrence
    lstm_rec_kernel<<<2 * NWG_DIR, 256, 0, stream>>>(xg_f, xg_b, w_hh_f, w_hh_b,
                                                     h_f, h_b, bar);

    // 4) output projection
    feats_kernel<<<Lseq, 224, 0, stream>>>(h_f, h_b, w_out, b_out, feats);

    // 5) Viterbi + backtrace -> d_out
    viterbi_kernel<<<1, 32, 0, stream>>>(feats, trans, bptrs, d_out);
}